// BWCaster_69260642615889
// MI455X (gfx1250) — compile-verified
//
#include <hip/hip_runtime.h>
#include <stdint.h>

typedef float v4f __attribute__((ext_vector_type(4)));

#define NJ   24
#define NPTS 131072
#define CCH  16
#define GRD  128
#define LSTRIDE 20   // padded floats per z-row in LDS (80B: 16B-aligned, bank-spread)

// workspace layout (floats)
#define PLANE_T_ELEMS ((size_t)NJ * 3 * GRD * GRD * CCH)  // 18,874,368
#define LINE_T_ELEMS  ((size_t)NJ * 3 * GRD * CCH)        //    147,456

__device__ __forceinline__ int imin(int a, int b) { return a < b ? a : b; }

// ---------------- CDNA5 async global->LDS helpers ----------------
__device__ __forceinline__ void async_copy_b128(uint32_t lds_byte_addr, const void* gaddr) {
  // CDNA5 GLOBAL_LOAD_ASYNC_TO_LDS_B128: vdst = VGPR holding LDS byte address,
  // vaddr = 64-bit global address, tracked by ASYNCcnt.
  asm volatile("global_load_async_to_lds_b128 %0, %1, off"
               :: "v"(lds_byte_addr), "v"((uint64_t)(uintptr_t)gaddr)
               : "memory");
}

__device__ __forceinline__ void wait_asynccnt0() {
#if __has_builtin(__builtin_amdgcn_s_wait_asynccnt)
  __builtin_amdgcn_s_wait_asynccnt((unsigned short)0);
#else
  asm volatile("s_wait_asynccnt 0" ::: "memory");
#endif
}

// ---------------- per-component accumulate (channel-last layout) ----------------
__device__ __forceinline__ void comp_accum(v4f& acc,
                                           const float* __restrict__ pb,   // plane tile (H,W,C)
                                           const float* __restrict__ lb,   // LDS line (L, LSTRIDE)
                                           float cx, float cy, float cz) {
  float fx = (cx + 1.0f) * 0.5f * 127.0f;
  float fy = (cy + 1.0f) * 0.5f * 127.0f;
  float fz = (cz + 1.0f) * 0.5f * 127.0f;
  float x0f = fminf(fmaxf(floorf(fx), 0.0f), 127.0f);
  float y0f = fminf(fmaxf(floorf(fy), 0.0f), 127.0f);
  float z0f = fminf(fmaxf(floorf(fz), 0.0f), 127.0f);
  int x0 = (int)x0f, y0 = (int)y0f, z0 = (int)z0f;
  int x1 = imin(x0 + 1, 127), y1 = imin(y0 + 1, 127), z1 = imin(z0 + 1, 127);
  float wx = fx - x0f, wy = fy - y0f, wz = fz - z0f;
  float w00 = (1.0f - wx) * (1.0f - wy);
  float w01 = wx * (1.0f - wy);
  float w10 = (1.0f - wx) * wy;
  float w11 = wx * wy;
  float a0 = 1.0f - wz, a1 = wz;
  const float* p00 = pb + (size_t)(y0 * GRD + x0) * CCH;
  const float* p01 = pb + (size_t)(y0 * GRD + x1) * CCH;
  const float* p10 = pb + (size_t)(y1 * GRD + x0) * CCH;
  const float* p11 = pb + (size_t)(y1 * GRD + x1) * CCH;
  const float* L0  = lb + z0 * LSTRIDE;
  const float* L1  = lb + z1 * LSTRIDE;
#pragma unroll
  for (int g = 0; g < 4; ++g) {
    v4f v00 = *(const v4f*)(p00 + 4 * g);
    v4f v01 = *(const v4f*)(p01 + 4 * g);
    v4f v10 = *(const v4f*)(p10 + 4 * g);
    v4f v11 = *(const v4f*)(p11 + 4 * g);
    v4f p = v00 * w00 + v01 * w01 + v10 * w10 + v11 * w11;
    v4f l = (*(const v4f*)(L0 + 4 * g)) * a0 + (*(const v4f*)(L1 + 4 * g)) * a1;
    acc += p * l;
  }
}

// ---------------- fused sampler (transposed tables) ----------------
#define PPT 4
__global__ __launch_bounds__(256) void k_fused(const float* __restrict__ xyz,
                                               const float* __restrict__ wsP,
                                               const float* __restrict__ wsL,
                                               float* __restrict__ out) {
  __shared__ float sL[3 * GRD * LSTRIDE];   // 30 KB, padded (z-stride 80B)
  const int j = blockIdx.y;

  // Stage all 3 line tables for this j via CDNA5 async global->LDS.
  // 3*128 z-rows * 4 chunks of 16B = 1536 chunks; 6 per thread.
  const float* lsrc = wsL + (size_t)j * 3 * GRD * CCH;   // (comp,z,c) contiguous
#pragma unroll
  for (int i = 0; i < 6; ++i) {
    int k  = threadIdx.x + i * 256;   // chunk id
    int zz = k >> 2;                  // comp*128+z
    int g  = k & 3;                   // 16B chunk within the 16 channels
    async_copy_b128((uint32_t)(uintptr_t)&sL[zz * LSTRIDE + 4 * g],
                    lsrc + (size_t)zz * CCH + 4 * g);
  }
  wait_asynccnt0();
  __syncthreads();

  const float* pbase = wsP + (size_t)j * 3 * GRD * GRD * CCH;
  const float* pb0 = pbase;
  const float* pb1 = pbase + (size_t)GRD * GRD * CCH;
  const float* pb2 = pbase + (size_t)2 * GRD * GRD * CCH;

  int n0 = blockIdx.x * (256 * PPT) + threadIdx.x;
#pragma unroll
  for (int p = 0; p < PPT; ++p) {
    int n = n0 + p * 256;
    size_t pidx = (size_t)j * NPTS + n;
    float c0 = xyz[pidx * 3 + 0];
    float c1 = xyz[pidx * 3 + 1];
    float c2 = xyz[pidx * 3 + 2];
    v4f acc = {0.0f, 0.0f, 0.0f, 0.0f};
    // MAT=((0,1),(0,2),(1,2)), VEC=(2,1,0)
    comp_accum(acc, pb0, sL + 0 * GRD * LSTRIDE, c0, c1, c2);
    comp_accum(acc, pb1, sL + 1 * GRD * LSTRIDE, c0, c2, c1);
    comp_accum(acc, pb2, sL + 2 * GRD * LSTRIDE, c1, c2, c0);
    float feat = acc.x + acc.y + acc.z + acc.w;
    out[pidx] = fmaxf(feat, 0.0f);
  }
}

// ---------------- transpose planes (J,C,H,W) -> (J,comp,H,W,C) ----------------
__global__ __launch_bounds__(256) void k_transpose_planes(
    const float* __restrict__ p0, const float* __restrict__ p1,
    const float* __restrict__ p2, float* __restrict__ dst) {
  int idx = blockIdx.x * 256 + threadIdx.x;   // (j,comp,y,x)
  int x    = idx & (GRD - 1);
  int y    = (idx >> 7) & (GRD - 1);
  int rest = idx >> 14;
  int comp = rest % 3;
  int j    = rest / 3;
  if (j >= NJ) return;
  const float* src = (comp == 0) ? p0 : (comp == 1) ? p1 : p2;
  const float* s = src + (size_t)j * CCH * GRD * GRD + (size_t)y * GRD + x;
  float v[CCH];
#pragma unroll
  for (int c = 0; c < CCH; ++c) v[c] = s[(size_t)c * GRD * GRD];
  float* d = dst + (((size_t)(j * 3 + comp) * GRD + y) * GRD + x) * CCH;
#pragma unroll
  for (int g = 0; g < 4; ++g) {
    v4f t = { v[4*g+0], v[4*g+1], v[4*g+2], v[4*g+3] };
    *(v4f*)(d + 4 * g) = t;
  }
}

// ---------------- transpose lines (J,C,L) -> (J,comp,L,C) ----------------
__global__ __launch_bounds__(256) void k_transpose_lines(
    const float* __restrict__ l0, const float* __restrict__ l1,
    const float* __restrict__ l2, float* __restrict__ dst) {
  int idx = blockIdx.x * 256 + threadIdx.x;   // (j,comp,z)
  int z    = idx & (GRD - 1);
  int rest = idx >> 7;
  int comp = rest % 3;
  int j    = rest / 3;
  if (j >= NJ) return;
  const float* src = (comp == 0) ? l0 : (comp == 1) ? l1 : l2;
  const float* s = src + (size_t)j * CCH * GRD + z;
  float* d = dst + ((size_t)(j * 3 + comp) * GRD + z) * CCH;
#pragma unroll
  for (int g = 0; g < 4; ++g) {
    v4f t = { s[(size_t)(4*g+0) * GRD], s[(size_t)(4*g+1) * GRD],
              s[(size_t)(4*g+2) * GRD], s[(size_t)(4*g+3) * GRD] };
    *(v4f*)(d + 4 * g) = t;
  }
}

// ---------------- fallback: direct gathers from original layout ----------------
__global__ __launch_bounds__(256) void k_fallback(
    const float* __restrict__ xyz,
    const float* __restrict__ p0, const float* __restrict__ p1, const float* __restrict__ p2,
    const float* __restrict__ l0, const float* __restrict__ l1, const float* __restrict__ l2,
    float* __restrict__ out) {
  size_t idx = (size_t)blockIdx.x * 256 + threadIdx.x;
  if (idx >= (size_t)NJ * NPTS) return;
  int j = (int)(idx / NPTS);
  float c0 = xyz[idx * 3 + 0];
  float c1 = xyz[idx * 3 + 1];
  float c2 = xyz[idx * 3 + 2];
  const float* P[3] = {p0, p1, p2};
  const float* L[3] = {l0, l1, l2};
  float cmx[3] = {c0, c0, c1};
  float cmy[3] = {c1, c2, c2};
  float cvz[3] = {c2, c1, c0};
  float feat = 0.0f;
#pragma unroll
  for (int i = 0; i < 3; ++i) {
    float fx = (cmx[i] + 1.0f) * 0.5f * 127.0f;
    float fy = (cmy[i] + 1.0f) * 0.5f * 127.0f;
    float fz = (cvz[i] + 1.0f) * 0.5f * 127.0f;
    float x0f = fminf(fmaxf(floorf(fx), 0.0f), 127.0f);
    float y0f = fminf(fmaxf(floorf(fy), 0.0f), 127.0f);
    float z0f = fminf(fmaxf(floorf(fz), 0.0f), 127.0f);
    int x0 = (int)x0f, y0 = (int)y0f, z0 = (int)z0f;
    int x1 = imin(x0 + 1, 127), y1 = imin(y0 + 1, 127), z1 = imin(z0 + 1, 127);
    float wx = fx - x0f, wy = fy - y0f, wz = fz - z0f;
    float w00 = (1.0f - wx) * (1.0f - wy);
    float w01 = wx * (1.0f - wy);
    float w10 = (1.0f - wx) * wy;
    float w11 = wx * wy;
    float a0 = 1.0f - wz, a1 = wz;
    const float* pp = P[i] + (size_t)j * CCH * GRD * GRD;
    const float* ll = L[i] + (size_t)j * CCH * GRD;
#pragma unroll
    for (int c = 0; c < CCH; ++c) {
      const float* pc = pp + (size_t)c * GRD * GRD;
      float pv = pc[y0 * GRD + x0] * w00 + pc[y0 * GRD + x1] * w01 +
                 pc[y1 * GRD + x0] * w10 + pc[y1 * GRD + x1] * w11;
      const float* lc = ll + (size_t)c * GRD;
      float lv = lc[z0] * a0 + lc[z1] * a1;
      feat += pv * lv;
    }
  }
  out[idx] = fmaxf(feat, 0.0f);
}

extern "C" void kernel_launch(void* const* d_in, const int* in_sizes, int n_in,
                              void* d_out, int out_size, void* d_ws, size_t ws_size,
                              hipStream_t stream) {
  // setup_inputs() order: xyz, plane0, line0, plane1, line1, plane2, line2
  const float* xyz = (const float*)d_in[0];
  const float* p0  = (const float*)d_in[1];
  const float* l0  = (const float*)d_in[2];
  const float* p1  = (const float*)d_in[3];
  const float* l1  = (const float*)d_in[4];
  const float* p2  = (const float*)d_in[5];
  const float* l2  = (const float*)d_in[6];
  float* out = (float*)d_out;

  const size_t need_bytes = (PLANE_T_ELEMS + LINE_T_ELEMS) * sizeof(float);
  if (ws_size >= need_bytes && d_ws != nullptr) {
    float* wsP = (float*)d_ws;
    float* wsL = wsP + PLANE_T_ELEMS;
    // 24*3*128*128 / 256 = 4608 blocks ; 24*3*128 / 256 = 36 blocks
    k_transpose_planes<<<4608, 256, 0, stream>>>(p0, p1, p2, wsP);
    k_transpose_lines<<<36, 256, 0, stream>>>(l0, l1, l2, wsL);
    dim3 grid(NPTS / (256 * PPT), NJ);   // (128, 24)
    k_fused<<<grid, 256, 0, stream>>>(xyz, wsP, wsL, out);
  } else {
    int blocks = (int)(((size_t)NJ * NPTS + 255) / 256);   // 12288
    k_fallback<<<blocks, 256, 0, stream>>>(xyz, p0, p1, p2, l0, l1, l2, out);
  }
}